// GemRelationLoss_31413390803558
// MI455X (gfx1250) — compile-verified
//
#include <hip/hip_runtime.h>
#include <hip/hip_bf16.h>

// Problem constants (from reference): logits [N,C,H,W] f32, params [N,J,3] i32.
#define N_ 16
#define C_ 8
#define H_ 512
#define W_ 1024
#define J_ 8
#define ROWS (N_ * C_ * H_)      // 65536 softmax rows
#define ROWS_PER_WAVE 16
#define WAVES_PER_BLOCK 8
#define ROWS_PER_BLOCK (ROWS_PER_WAVE * WAVES_PER_BLOCK)  // 128

typedef float v2f __attribute__((ext_vector_type(2)));
typedef float v8f __attribute__((ext_vector_type(8)));

// ---------------------------------------------------------------------------
// Kernel 1: pos[n,c,h] = sum_w(w * e^x) / sum_w(e^x)  via f32 WMMA.
// One wave handles a 16-row tile: E[16x1024] @ B[1024x2] with
// B[:,0]=1, B[:,1]=w, accumulated with V_WMMA_F32_16X16X4_F32 (K=4 chunks).
// A-fragment layout (ISA 7.12.2, 32-bit A 16x4): lane L<16 holds
// A[M=L,K=0..1] in {v0,v1}; lane L+16 holds A[M=L,K=2..3].
// B 4x16: lane n<16 holds B[K=0..1, n]; lane n+16 holds B[K=2..3, n].
// ---------------------------------------------------------------------------
__global__ __launch_bounds__(256) void k_pos(const float* __restrict__ logits,
                                             float* __restrict__ pos) {
  const int lane = threadIdx.x & 31;
  const int wave = threadIdx.x >> 5;
  const int rowBase = blockIdx.x * ROWS_PER_BLOCK + wave * ROWS_PER_WAVE;
  const int r = lane & 15;            // tile row for A; column index n for B
  const int kofs = (lane >> 4) << 1;  // 0 (low half-wave) or 2 (high half-wave)

  const float* __restrict__ src = logits + (size_t)(rowBase + r) * W_ + kofs;
  const bool isN0 = (r == 0);         // B column 0: ones (-> sum e)
  const bool isN1 = (r == 1);         // B column 1: w     (-> sum w*e)

  v8f c = {0.f, 0.f, 0.f, 0.f, 0.f, 0.f, 0.f, 0.f};
  float kw = (float)kofs;             // w-index of this lane's first element

  for (int k = 0; k < W_; k += 4) {
    v2f x = *(const v2f*)(src + k);   // 8B load: two consecutive row elements
    v2f a;
    a.x = __expf(x.x);
    a.y = __expf(x.y);
    v2f b;
    b.x = isN0 ? 1.0f : (isN1 ? kw : 0.0f);
    b.y = isN0 ? 1.0f : (isN1 ? (kw + 1.0f) : 0.0f);
    // D = A*B + C; args: (neg_a, A, neg_b, B, c_mod, C, reuse_a, reuse_b)
    c = __builtin_amdgcn_wmma_f32_16x16x4_f32(false, a, false, b, (short)0, c,
                                              false, false);
    kw += 4.0f;
  }

  // D layout (32-bit C/D 16x16): VGPR i, lanes 0-15 -> M=i, N=lane;
  // lanes 16-31 -> M=i+8, N=lane-16.  We need columns N=0 (sumE), N=1 (sumW).
  __shared__ float sE[WAVES_PER_BLOCK][16];
  __shared__ float sW[WAVES_PER_BLOCK][16];
  if (lane == 0) {
    for (int i = 0; i < 8; ++i) sE[wave][i] = c[i];
  } else if (lane == 16) {
    for (int i = 0; i < 8; ++i) sE[wave][8 + i] = c[i];
  } else if (lane == 1) {
    for (int i = 0; i < 8; ++i) sW[wave][i] = c[i];
  } else if (lane == 17) {
    for (int i = 0; i < 8; ++i) sW[wave][8 + i] = c[i];
  }
  __syncthreads();
  if (lane < 16) {
    pos[rowBase + lane] = sW[wave][lane] / sE[wave][lane];
  }
}

// ---------------------------------------------------------------------------
// Kernel 2: per-(n,j) masked smooth-L1 over row diffs of the gathered channel.
// One block per (n,j); fixed-order LDS tree reduction -> deterministic.
// ---------------------------------------------------------------------------
__global__ __launch_bounds__(256) void k_loss(const float* __restrict__ pos,
                                              const int* __restrict__ params,
                                              float* __restrict__ partials) {
  const int tid = threadIdx.x;
  const int n = blockIdx.x / J_;
  const int j = blockIdx.x % J_;
  const int* p = params + ((size_t)n * J_ + j) * 3;
  const int start = p[0];
  const int end = p[1];
  const int ch = p[2];
  const float* __restrict__ row = pos + ((size_t)n * C_ + ch) * H_;

  float s = 0.0f, cnt = 0.0f;
  for (int z = tid; z < H_ - 1; z += 256) {
    if (z >= start && z < end) {
      float d = row[z] - row[z + 1];
      float x = fabsf(d);
      s += (x < 1.0f) ? 0.5f * d * d : (x - 0.5f);
      cnt += 1.0f;
    }
  }

  __shared__ float ls[256];
  __shared__ float lc[256];
  ls[tid] = s;
  lc[tid] = cnt;
  __syncthreads();
  for (int off = 128; off > 0; off >>= 1) {
    if (tid < off) {
      ls[tid] += ls[tid + off];
      lc[tid] += lc[tid + off];
    }
    __syncthreads();
  }
  if (tid == 0) {
    partials[2 * blockIdx.x] = ls[0];
    partials[2 * blockIdx.x + 1] = lc[0];
  }
}

// ---------------------------------------------------------------------------
// Kernel 3: reduce the 128 partial (sum,count) pairs in fixed order; emit mean.
// ---------------------------------------------------------------------------
__global__ __launch_bounds__(128) void k_final(const float* __restrict__ partials,
                                               float* __restrict__ out,
                                               int nparts) {
  const int tid = threadIdx.x;
  float s = 0.0f, cnt = 0.0f;
  for (int i = tid; i < nparts; i += 128) {
    s += partials[2 * i];
    cnt += partials[2 * i + 1];
  }
  __shared__ float ls[128];
  __shared__ float lc[128];
  ls[tid] = s;
  lc[tid] = cnt;
  __syncthreads();
  for (int off = 64; off > 0; off >>= 1) {
    if (tid < off) {
      ls[tid] += ls[tid + off];
      lc[tid] += lc[tid + off];
    }
    __syncthreads();
  }
  if (tid == 0) out[0] = ls[0] / lc[0];
}

extern "C" void kernel_launch(void* const* d_in, const int* in_sizes, int n_in,
                              void* d_out, int out_size, void* d_ws,
                              size_t ws_size, hipStream_t stream) {
  const float* logits = (const float*)d_in[0];  // [N,C,H,W] f32
  const int* params = (const int*)d_in[1];      // [N,J,3] i32
  float* out = (float*)d_out;                   // scalar f32

  float* pos = (float*)d_ws;                    // ROWS floats (256 KB)
  float* partials = pos + ROWS;                 // 2 * N*J floats

  k_pos<<<ROWS / ROWS_PER_BLOCK, 256, 0, stream>>>(logits, pos);
  k_loss<<<N_ * J_, 256, 0, stream>>>(pos, params, partials);
  k_final<<<1, 128, 0, stream>>>(partials, out, N_ * J_);
}